// GraphSAGE_14010183320060
// MI455X (gfx1250) — compile-verified
//
#include <hip/hip_runtime.h>

typedef float v2f __attribute__((ext_vector_type(2)));
typedef float v8f __attribute__((ext_vector_type(8)));

// ---------------------------------------------------------------------------
// Zero-fill fp32 scratch (agg1 | agg2 | cnt are contiguous in d_ws).
// ---------------------------------------------------------------------------
__global__ void sage_zero_f32(float* __restrict__ p, long long n) {
  long long i = (long long)blockIdx.x * blockDim.x + threadIdx.x;
  if (i < n) p[i] = 0.0f;
}

// ---------------------------------------------------------------------------
// Degree count: cnt[dst] += 1  (fp32 so the mean divide stays in fp32).
// ---------------------------------------------------------------------------
__global__ void sage_degree(const long long* __restrict__ ei, int E,
                            float* __restrict__ cnt) {
  int e = blockIdx.x * blockDim.x + threadIdx.x;
  if (e < E) atomicAdd(cnt + (int)ei[(long long)E + e], 1.0f);
}

// ---------------------------------------------------------------------------
// Edge scatter: agg[dst, :] += feat[src, :].  One wave32 per edge, each lane
// handles D/32 features; loads and atomics are lane-coalesced.  agg lives in
// d_ws and fits in the 192MB L2, so atomics resolve at L2 bandwidth.
// ---------------------------------------------------------------------------
template <int D>
__global__ void sage_scatter_add(const long long* __restrict__ ei, int E,
                                 const float* __restrict__ feat,
                                 float* __restrict__ agg) {
  int gid  = blockIdx.x * blockDim.x + threadIdx.x;
  int wave = gid >> 5;
  int lane = gid & 31;
  if (wave >= E) return;
  int src = (int)ei[wave];                   // edge_index[0][e]
  int dst = (int)ei[(long long)E + wave];    // edge_index[1][e]
  const float* fs = feat + (long long)src * D;
  float*       ad = agg  + (long long)dst * D;
#pragma unroll
  for (int i = 0; i < D / 32; ++i)
    atomicAdd(ad + lane + 32 * i, fs[lane + 32 * i]);
}

// ---------------------------------------------------------------------------
// Fused SAGEConv GEMM:  out = act( (agg/max(cnt,1)) @ Wl + xin @ Wr + bias ).
// One wave32 owns one 16x16 output tile; K accumulated with
// v_wmma_f32_16x16x4_f32 (the fp32 matrix path on CDNA5).
//
// ISA fragment layouts (wave32):
//   A 16x4 f32 : lane%16 = row M, lane/16 selects K-pair -> float2 load.
//   B 4x16 f32 : lane%16 = col N, K = 2*(lane/16) + vgpr.
//   C/D        : VGPR r -> (M = r + 8*(lane/16), N = lane%16).
// ---------------------------------------------------------------------------
template <int K, int NO, bool RELU>
__global__ void sage_wmma_gemm(const float* __restrict__ agg,
                               const float* __restrict__ cnt,
                               const float* __restrict__ xin,
                               const float* __restrict__ Wl,
                               const float* __restrict__ Wr,
                               const float* __restrict__ bias,
                               float* __restrict__ out, int nNodes) {
  constexpr int NT = NO / 16;
  int gid   = blockIdx.x * blockDim.x + threadIdx.x;
  int wave  = gid >> 5;
  int lane  = gid & 31;
  int tiles = (nNodes / 16) * NT;
  if (wave >= tiles) return;  // wave-uniform: EXEC all-ones at every WMMA

  int tm   = wave / NT;
  int tn   = wave % NT;
  int half = lane >> 4;   // 0: K {0,1} / rows M 0..7   1: K {2,3} / rows M 8..15
  int l16  = lane & 15;

  int rowA = tm * 16 + l16;   // A-fragment row for this lane
  int colB = tn * 16 + l16;   // B/C column for this lane

  float invc = 1.0f / fmaxf(cnt[rowA], 1.0f);   // mean divisor (isolated -> /1)
  float bval = bias[colB];

  v8f acc;
#pragma unroll
  for (int r = 0; r < 8; ++r) acc[r] = bval;    // bias pre-loaded into C

  const float* arow = agg + (long long)rowA * K;
  const float* xrow = xin + (long long)rowA * K;

  // mean @ Wl  --- scale the A fragment by 1/deg on the fly
#pragma unroll
  for (int k = 0; k < K; k += 4) {
    v2f a = *(const v2f*)(arow + k + 2 * half);
    a.x *= invc;
    a.y *= invc;
    v2f b;
    b.x = Wl[(k + 2 * half + 0) * NO + colB];
    b.y = Wl[(k + 2 * half + 1) * NO + colB];
    acc = __builtin_amdgcn_wmma_f32_16x16x4_f32(false, a, false, b, (short)0,
                                                acc, false, false);
  }

  // x @ Wr
#pragma unroll
  for (int k = 0; k < K; k += 4) {
    v2f a = *(const v2f*)(xrow + k + 2 * half);
    v2f b;
    b.x = Wr[(k + 2 * half + 0) * NO + colB];
    b.y = Wr[(k + 2 * half + 1) * NO + colB];
    acc = __builtin_amdgcn_wmma_f32_16x16x4_f32(false, a, false, b, (short)0,
                                                acc, false, false);
  }

  float* orow = out + (long long)tm * 16 * NO + colB;
#pragma unroll
  for (int r = 0; r < 8; ++r) {
    float v = acc[r];
    if (RELU) v = fmaxf(v, 0.0f);
    orow[(long long)(r + 8 * half) * NO] = v;
  }
}

// ---------------------------------------------------------------------------
// Host-side orchestration (all on `stream`, graph-capture safe).
// Input order: x, W1l, W1r, b1, W2l, W2r, b2, edge_index(int64, [2,E])
// Workspace layout (floats): agg1[N*64] | agg2[N*128] | cnt[N] | h[N*128]
// ---------------------------------------------------------------------------
extern "C" void kernel_launch(void* const* d_in, const int* in_sizes, int n_in,
                              void* d_out, int out_size, void* d_ws,
                              size_t ws_size, hipStream_t stream) {
  const float*     x   = (const float*)d_in[0];
  const float*     W1l = (const float*)d_in[1];
  const float*     W1r = (const float*)d_in[2];
  const float*     b1  = (const float*)d_in[3];
  const float*     W2l = (const float*)d_in[4];
  const float*     W2r = (const float*)d_in[5];
  const float*     b2  = (const float*)d_in[6];
  const long long* ei  = (const long long*)d_in[7];

  const int N = in_sizes[0] / 64;  // 100000 (divisible by 16)
  const int E = in_sizes[7] / 2;   // 1000000

  float* agg1 = (float*)d_ws;                  // N*64
  float* agg2 = agg1 + (long long)N * 64;      // N*128
  float* cnt  = agg2 + (long long)N * 128;     // N
  float* h    = cnt + N;                       // N*128 (fully overwritten)

  // 1) zero agg1|agg2|cnt (contiguous)
  long long zn = (long long)N * (64 + 128 + 1);
  sage_zero_f32<<<(int)((zn + 255) / 256), 256, 0, stream>>>(agg1, zn);

  // 2) degrees (shared by both layers)
  sage_degree<<<(E + 255) / 256, 256, 0, stream>>>(ei, E, cnt);

  // 3) layer-1 neighbor sum
  sage_scatter_add<64><<<(int)(((long long)E * 32 + 255) / 256), 256, 0,
                         stream>>>(ei, E, x, agg1);

  // 4) h = relu(mean1 @ W1l + x @ W1r + b1)   [N,128]
  {
    int waves = (N / 16) * (128 / 16);
    sage_wmma_gemm<64, 128, true>
        <<<(int)(((long long)waves * 32 + 255) / 256), 256, 0, stream>>>(
            agg1, cnt, x, W1l, W1r, b1, h, N);
  }

  // 5) layer-2 neighbor sum over h
  sage_scatter_add<128><<<(int)(((long long)E * 32 + 255) / 256), 256, 0,
                          stream>>>(ei, E, h, agg2);

  // 6) out = mean2 @ W2l + h @ W2r + b2       [N,64]
  {
    int waves = (N / 16) * (64 / 16);
    sage_wmma_gemm<128, 64, false>
        <<<(int)(((long long)waves * 32 + 255) / 256), 256, 0, stream>>>(
            agg2, cnt, h, W2l, W2r, b2, (float*)d_out, N);
  }
}